// LLSA_63307817943524
// MI455X (gfx1250) — compile-verified
//
#include <hip/hip_runtime.h>

typedef __attribute__((ext_vector_type(2))) float v2f;
typedef __attribute__((ext_vector_type(8))) float v8f;

#define B_DIM 256
#define T_DIM 16384
#define WIN   5
#define HID   5
#define N_OUT (T_DIM - WIN)   /* 16379 */

// One wave (32 lanes) owns a tile of 16 consecutive n values and loops over all
// 256 series rows (b). Per b:
//   - one coalesced global_load_b32: v = s[b, nbase+lane]  (lane 0..31 covers the
//     21 floats the tile needs; clamped at T-1, clamped lanes only feed n >= N rows)
//   - A-fragments built via ds_bpermute shuffles in the ISA 16x4 f32 A layout
//   - e(16x16) = A1 x B1 + (A2 x B2 + 0) via two chained v_wmma_f32_16x16x4_f32
//   - acc[r] += e[r]^2  (8 VGPRs per lane)
// After the b loop, losses[n] = (sum of acc over h-lanes 0..4 / 16..20) / (B*H).
__global__ __launch_bounds__(256) void llsa_losses_kernel(
    const float* __restrict__ series,
    const float* __restrict__ weight,
    const float* __restrict__ bias,
    float* __restrict__ losses)
{
    const int lane  = threadIdx.x & 31;
    const int wave  = threadIdx.x >> 5;
    const int nbase = blockIdx.x * 128 + wave * 16;   // 8 waves * 16 n per block

    const int  hcol = lane & 15;
    const bool lo   = (lane < 16);

    // G[k][h], k=0..7 (K dim), h=0..15 (cols, only 0..4 meaningful):
    //   k<5 : W[h][k]          (the window weights)
    //   k==h+1 (1<=k<=5): -1   (subtracts target s[n+1+h])
    //   k==6 : bias[h]         (x[6]==1 constant column)
    //   k==7 : 0               (pad)
    auto gval = [&](int k) -> float {
        if (hcol >= HID) return 0.0f;
        float g = 0.0f;
        if (k < WIN) g = weight[hcol * WIN + k];
        if (k == 6)  g = bias[hcol];
        if (k >= 1 && k == hcol + 1) g -= 1.0f;
        return g;
    };

    // B-matrix (4x16) WMMA layout (mirror of documented A 16x4 layout):
    // VGPR0: lanes 0-15 = row K=0, lanes 16-31 = row K=2; VGPR1: K=1 / K=3.
    v2f b1, b2;
    b1.x = gval(lo ? 0 : 2);
    b1.y = gval(lo ? 1 : 3);
    b2.x = gval(lo ? 4 : 6);
    b2.y = gval(lo ? 5 : 7);

    // A 16x4 f32 layout: lanes 0-15 hold M=lane with VGPR0=K0, VGPR1=K1;
    // lanes 16-31 hold M=lane-16 with VGPR0=K2, VGPR1=K3.
    // Second WMMA covers k=0..3 -> A1[m][k] = s[nbase+m+k]
    // First covers k=4..7    -> A2: lo lanes s[nbase+m+4/5], hi lanes 1.0/0.0
    const int selLo   = (lane & 15) + (lo ? 0 : 2);
    const int sel4    = (lane & 15) + 4;
    const int idxLoad = min(nbase + lane, T_DIM - 1);

    v8f acc = {0.f, 0.f, 0.f, 0.f, 0.f, 0.f, 0.f, 0.f};

    for (int b = 0; b < B_DIM; ++b) {
        const float v = series[(size_t)b * T_DIM + idxLoad];

        v2f a1, a2;
        a1.x = __shfl(v, selLo,     32);
        a1.y = __shfl(v, selLo + 1, 32);
        const float s4 = __shfl(v, sel4,     32);
        const float s5 = __shfl(v, sel4 + 1, 32);
        a2.x = lo ? s4 : 1.0f;   // hi half = K6 column (constant 1 for bias)
        a2.y = lo ? s5 : 0.0f;   // hi half = K7 column (zero pad)

        v8f c0 = {0.f, 0.f, 0.f, 0.f, 0.f, 0.f, 0.f, 0.f};
        v8f e = __builtin_amdgcn_wmma_f32_16x16x4_f32(
                    false, a2, false, b2, (short)0, c0, false, false);
        e = __builtin_amdgcn_wmma_f32_16x16x4_f32(
                    false, a1, false, b1, (short)0, e, false, false);

#pragma unroll
        for (int r = 0; r < 8; ++r)
            acc[r] = fmaf(e[r], e[r], acc[r]);
    }

    // D layout: VGPR r holds row M=r (lanes 0-15, col=lane) and row M=r+8
    // (lanes 16-31, col=lane-16). Reduce the 5 used h-columns per row.
    const float inv = 1.0f / (float)(B_DIM * HID);
#pragma unroll
    for (int r = 0; r < 8; ++r) {
        float slo = 0.0f, shi = 0.0f;
#pragma unroll
        for (int hh = 0; hh < HID; ++hh) {
            slo += __shfl(acc[r], hh,      32);
            shi += __shfl(acc[r], 16 + hh, 32);
        }
        if (lane == 0) {
            const int nlo = nbase + r;
            const int nhi = nbase + 8 + r;
            if (nlo < N_OUT) losses[nlo] = slo * inv;
            if (nhi < N_OUT) losses[nhi] = shi * inv;
        }
    }
}

__global__ void llsa_diffs_kernel(const float* __restrict__ losses,
                                  const float* __restrict__ thr_p,
                                  float* __restrict__ diffs,
                                  float* __restrict__ mask)
{
    const int i = blockIdx.x * blockDim.x + threadIdx.x;
    if (i < N_OUT - 1) {
        const float d = fabsf(losses[i + 1] - losses[i]);
        diffs[i] = d;
        mask[i]  = (d > thr_p[0]) ? 1.0f : 0.0f;
    }
}

extern "C" void kernel_launch(void* const* d_in, const int* in_sizes, int n_in,
                              void* d_out, int out_size, void* d_ws, size_t ws_size,
                              hipStream_t stream)
{
    (void)in_sizes; (void)n_in; (void)d_ws; (void)ws_size; (void)out_size;

    const float* series = (const float*)d_in[0];   // (256, 16384) f32
    const float* weight = (const float*)d_in[1];   // (5, 5) f32
    const float* bias   = (const float*)d_in[2];   // (5,) f32
    const float* thr    = (const float*)d_in[3];   // scalar f32

    float* out    = (float*)d_out;
    float* losses = out;                        // [N_OUT]
    float* diffs  = out + N_OUT;                // [N_OUT-1]
    float* mask   = out + N_OUT + (N_OUT - 1);  // [N_OUT-1] as 1.0/0.0

    // 8 waves/block * 16 n per wave = 128 n per block
    const int blocks1 = (N_OUT + 127) / 128;    // 128
    llsa_losses_kernel<<<blocks1, 256, 0, stream>>>(series, weight, bias, losses);

    const int blocks2 = (N_OUT - 1 + 255) / 256;
    llsa_diffs_kernel<<<blocks2, 256, 0, stream>>>(losses, thr, diffs, mask);
}